// EncoderPp_11991548690861
// MI455X (gfx1250) — compile-verified
//
#include <hip/hip_runtime.h>
#include <hip/hip_bf16.h>
#include <math.h>

// ---------------------------------------------------------------------------
// PointNet++-style encoder for MI455X (gfx1250, wave32).
//   1) hidden = tanh(x @ W1 + b1)                         (elementwise kernel)
//   2) lf = tanh(hidden @ W2 + b2)                        (f32 WMMA GEMM)
//   3) FPS on x -> ctr1 (m=2048); FPS on ctr1 -> ctr2 (m=512)
//   4) t1 = [lf, zone, pos] @ c1_W                        (f32 WMMA GEMM, K=67)
//   5) h1[i] = max_{ball(ctr1_i, 0.5)} t1[j] + (c1_b - ctr1_i @ c1_W[65:67])
//   6) t2 = [h1, ctr1] @ c2_W                             (f32 WMMA GEMM, K=66)
//   7) h2[i] = max_{ball(ctr2_i, 1.0)} t2[j] + (c2_b - ctr2_i @ c2_W[64:66])
//   8) gf = max_rows([h2, ctr2] @ c3_W + c3_b)            (f32 WMMA + atomicMax)
// ---------------------------------------------------------------------------

typedef __attribute__((ext_vector_type(2))) float v2f;
typedef __attribute__((ext_vector_type(8))) float v8f;

#define BATCH 4
#define NPTS  4096
#define M1    2048
#define M2    512
#define GF    1024

// ---- order-preserving float<->uint encoding for atomic max ----------------
__device__ __forceinline__ unsigned f2ord(float f) {
    unsigned u = __float_as_uint(f);
    return (u & 0x80000000u) ? ~u : (u | 0x80000000u);
}
__device__ __forceinline__ float ord2f(unsigned k) {
    return (k & 0x80000000u) ? __uint_as_float(k & 0x7fffffffu)
                             : __uint_as_float(~k);
}

// ---------------------------------------------------------------------------
// Kernel 1: hidden = tanh(x @ W1 + b1), x:[M,2], W1:[2,64] -> hidden:[M,64]
// ---------------------------------------------------------------------------
__global__ void hidden_kernel(const float* __restrict__ x,
                              const float* __restrict__ W1,
                              const float* __restrict__ b1,
                              float* __restrict__ hidden, int M) {
    int idx = blockIdx.x * blockDim.x + threadIdx.x;
    if (idx >= M * 64) return;
    int r = idx >> 6, f = idx & 63;
    float v = x[r * 2 + 0] * W1[f] + x[r * 2 + 1] * W1[64 + f] + b1[f];
    hidden[idx] = tanhf(v);
}

// ---------------------------------------------------------------------------
// Compile-time aux-column fetch (branchless at runtime).
// ---------------------------------------------------------------------------
template <int KREL, int C1, int C2>
__device__ __forceinline__ float auxload(const float* __restrict__ A1,
                                         const float* __restrict__ A2,
                                         int row) {
    if constexpr (KREL < C1) {
        return A1[row * C1 + KREL];
    } else if constexpr (KREL - C1 < C2) {
        return A2[row * C2 + (KREL - C1)];
    } else {
        return 0.0f;
    }
}

// ---------------------------------------------------------------------------
// f32 WMMA GEMM: T = [A0(C0) | A1(C1) | A2(C2) | 0-pad] @ W,  K = KTOT.
// One wave per 16-row x 64-col tile; V_WMMA_F32_16X16X4_F32, fully unrolled,
// branchless A/B fetch. A layout (ISA 7.12.2 16x4 f32): lanes 0-15 = rows,
// VGPR0 = {K0 | K2 per half}, VGPR1 = {K1 | K3}; B mirrors C striping.
// OUTMODE: 0 raw store, 1 tanh(acc+bias), 2 per-batch atomicMax(acc+bias).
// ---------------------------------------------------------------------------
template <int OUTMODE, int C0, int C1, int C2, int KTOT>
__global__ void gemm_wmma_kernel(const float* __restrict__ A0,
                                 const float* __restrict__ A1,
                                 const float* __restrict__ A2,
                                 const float* __restrict__ W, int F,
                                 const float* __restrict__ bias,
                                 float* __restrict__ out,
                                 unsigned* __restrict__ gfmax,
                                 int rowsPerBatch) {
    constexpr int KP = (KTOT + 3) & ~3;
    static_assert((C0 & 3) == 0, "C0 multiple of 4");
    static_assert(KP - C0 == 0 || KP - C0 == 4, "at most one tail K-step");

    int lane = threadIdx.x & 31;
    int half = lane >> 4;       // 0: lanes 0-15, 1: lanes 16-31
    int mrow = lane & 15;
    int tm = blockIdx.x;        // 16-row tile
    int colBase = blockIdx.y * 64;
    int arow = tm * 16 + mrow;  // A row this lane fetches

    v8f c[4];
#pragma unroll
    for (int q = 0; q < 4; ++q)
#pragma unroll
        for (int v = 0; v < 8; ++v) c[q][v] = 0.0f;

    const float* a0row = A0 + (size_t)arow * C0;

    // ---- main K loop over A0 columns: branchless, 8B A loads ----
#pragma unroll
    for (int k0 = 0; k0 < C0; k0 += 4) {
        int ka = k0 + 2 * half;                 // even -> 8B aligned
        v2f a = *(const v2f*)(a0row + ka);
#pragma unroll
        for (int nt = 0; nt < 4; ++nt) {
            int n = colBase + nt * 16 + mrow;
            v2f bm;
            bm.x = W[ka * F + n];
            bm.y = W[(ka + 1) * F + n];
            c[nt] = __builtin_amdgcn_wmma_f32_16x16x4_f32(
                false, a, false, bm, (short)0, c[nt], false, false);
        }
    }

    // ---- single tail K-step covering aux columns + zero pad ----
    if constexpr (KP > C0) {
        float t0 = auxload<0, C1, C2>(A1, A2, arow);
        float t1 = auxload<1, C1, C2>(A1, A2, arow);
        float t2 = auxload<2, C1, C2>(A1, A2, arow);
        float t3 = auxload<3, C1, C2>(A1, A2, arow);
        v2f a;
        a.x = half ? t2 : t0;
        a.y = half ? t3 : t1;
        int ka = C0 + 2 * half;
        int kx = (ka < KTOT) ? ka : (KTOT - 1);          // clamp, select 0 below
        int ky = (ka + 1 < KTOT) ? (ka + 1) : (KTOT - 1);
#pragma unroll
        for (int nt = 0; nt < 4; ++nt) {
            int n = colBase + nt * 16 + mrow;
            float wx = W[kx * F + n];
            float wy = W[ky * F + n];
            v2f bm;
            bm.x = (ka < KTOT) ? wx : 0.0f;
            bm.y = (ka + 1 < KTOT) ? wy : 0.0f;
            c[nt] = __builtin_amdgcn_wmma_f32_16x16x4_f32(
                false, a, false, bm, (short)0, c[nt], false, false);
        }
    }

    // ---- epilogue ----
#pragma unroll
    for (int nt = 0; nt < 4; ++nt) {
        int n = colBase + nt * 16 + mrow;
#pragma unroll
        for (int v = 0; v < 8; ++v) {
            int m = tm * 16 + v + 8 * half;  // C layout: VGPR v -> rows v / v+8
            float val = c[nt][v];
            if constexpr (OUTMODE == 0) {
                out[(size_t)m * F + n] = val;
            } else if constexpr (OUTMODE == 1) {
                out[(size_t)m * F + n] = tanhf(val + bias[n]);
            } else {
                float g = val + bias[n];
                atomicMax(&gfmax[(m / rowsPerBatch) * F + n], f2ord(g));
            }
        }
    }
}

// ---------------------------------------------------------------------------
// Farthest point sampling. One block (256 threads) per batch.
// Matches jnp semantics: start at 0, argmax picks FIRST max index.
// ---------------------------------------------------------------------------
__global__ void fps_kernel(const float* __restrict__ pos, int n, int m,
                           float* __restrict__ ctr) {
    __shared__ float mind2[NPTS];
    __shared__ float rval[256];
    __shared__ int ridx[256];
    __shared__ int s_last;

    int b = blockIdx.x;
    int tid = threadIdx.x;
    const float* pb = pos + (size_t)b * n * 2;
    float* cb = ctr + (size_t)b * m * 2;

    for (int j = tid; j < n; j += 256) mind2[j] = 1e10f;
    if (tid == 0) {
        s_last = 0;
        cb[0] = pb[0];
        cb[1] = pb[1];
    }
    __syncthreads();

    int last = 0;
    for (int it = 1; it < m; ++it) {
        float lx = pb[last * 2 + 0];
        float ly = pb[last * 2 + 1];
        float bv = -1.0f;
        int bi = n;
        for (int j = tid; j < n; j += 256) {
            float2 p = ((const float2*)pb)[j];
            float dx = p.x - lx;
            float dy = p.y - ly;
            float d2 = dx * dx + dy * dy;
            float md = fminf(mind2[j], d2);
            mind2[j] = md;
            // strictly-greater keeps earliest index within this thread's set
            if (md > bv) { bv = md; bi = j; }
        }
        rval[tid] = bv;
        ridx[tid] = bi;
        __syncthreads();
        for (int s = 128; s > 0; s >>= 1) {
            if (tid < s) {
                float ov = rval[tid + s];
                int oi = ridx[tid + s];
                if (ov > rval[tid] || (ov == rval[tid] && oi < ridx[tid])) {
                    rval[tid] = ov;
                    ridx[tid] = oi;
                }
            }
            __syncthreads();
        }
        if (tid == 0) {
            int nxt = ridx[0];
            s_last = nxt;
            cb[it * 2 + 0] = pb[nxt * 2 + 0];
            cb[it * 2 + 1] = pb[nxt * 2 + 1];
        }
        __syncthreads();
        last = s_last;
    }
}

// ---------------------------------------------------------------------------
// Radius max-aggregation: one wave per center.
// h[b,i,f] = max_{j in ball} t[b,j,f] + bias[f] - cx*Wp[0,f] - cy*Wp[1,f]
// ---------------------------------------------------------------------------
template <int U>
__global__ void agg_kernel(const float* __restrict__ t,
                           const float* __restrict__ pos,
                           const float* __restrict__ ctr,
                           const float* __restrict__ Wp,
                           const float* __restrict__ bias,
                           float* __restrict__ out,
                           int n, int m, float r2) {
    const int F = U * 32;
    int gtid = blockIdx.x * blockDim.x + threadIdx.x;
    int w = gtid >> 5;
    int lane = threadIdx.x & 31;
    if (w >= BATCH * m) return;
    int b = w / m;
    int i = w - b * m;

    float2 cc = ((const float2*)ctr)[b * m + i];
    float cx = cc.x, cy = cc.y;

    float acc[U];
#pragma unroll
    for (int u = 0; u < U; ++u) acc[u] = -1e30f;

    const float2* pb = (const float2*)(pos + (size_t)b * n * 2);
    const float* tb = t + (size_t)b * n * F;
    for (int j = 0; j < n; ++j) {
        float2 p = pb[j];
        float dx = p.x - cx;
        float dy = p.y - cy;
        if (dx * dx + dy * dy <= r2) {   // wave-uniform branch
#pragma unroll
            for (int u = 0; u < U; ++u)
                acc[u] = fmaxf(acc[u], tb[(size_t)j * F + lane + 32 * u]);
        }
    }
#pragma unroll
    for (int u = 0; u < U; ++u) {
        int f = lane + 32 * u;
        out[(size_t)(b * m + i) * F + f] =
            acc[u] + bias[f] - cx * Wp[f] - cy * Wp[F + f];
    }
}

// ---------------------------------------------------------------------------
// gf_enc init + decode
// ---------------------------------------------------------------------------
__global__ void init_gf_kernel(unsigned* __restrict__ gf, int n) {
    int i = blockIdx.x * blockDim.x + threadIdx.x;
    if (i < n) gf[i] = 0u;  // below any real encoded value; always overwritten
}
__global__ void decode_gf_kernel(const unsigned* __restrict__ gf,
                                 float* __restrict__ out, int n) {
    int i = blockIdx.x * blockDim.x + threadIdx.x;
    if (i < n) out[i] = ord2f(gf[i]);
}

// ---------------------------------------------------------------------------
extern "C" void kernel_launch(void* const* d_in, const int* in_sizes, int n_in,
                              void* d_out, int out_size, void* d_ws, size_t ws_size,
                              hipStream_t stream) {
    const float* x     = (const float*)d_in[0];   // [B,N,2]
    const float* zones = (const float*)d_in[1];   // [B,N,1]
    const float* lf_W1 = (const float*)d_in[2];   // [2,64]
    const float* lf_b1 = (const float*)d_in[3];   // [64]
    const float* lf_W2 = (const float*)d_in[4];   // [64,64]
    const float* lf_b2 = (const float*)d_in[5];   // [64]
    const float* c1_W  = (const float*)d_in[6];   // [67,64]
    const float* c1_b  = (const float*)d_in[7];   // [64]
    const float* c2_W  = (const float*)d_in[8];   // [66,128]
    const float* c2_b  = (const float*)d_in[9];   // [128]
    const float* c3_W  = (const float*)d_in[10];  // [130,1024]
    const float* c3_b  = (const float*)d_in[11];  // [1024]

    float* out    = (float*)d_out;
    float* lf_out = out;                               // [B*N,64]
    float* gf_out = out + (size_t)BATCH * NPTS * 64;   // [B,1024]

    const int M = BATCH * NPTS;  // 16384

    // workspace layout (floats)
    float* ws = (float*)d_ws;
    size_t o = 0;
    float* hidden = ws + o;  o += (size_t)M * 64;           // 1,048,576
    float* t1     = ws + o;  o += (size_t)M * 64;           // 1,048,576
    float* ctr1   = ws + o;  o += (size_t)BATCH * M1 * 2;   //    16,384
    float* h1     = ws + o;  o += (size_t)BATCH * M1 * 64;  //   524,288
    float* t2     = ws + o;  o += (size_t)BATCH * M1 * 128; // 1,048,576
    float* ctr2   = ws + o;  o += (size_t)BATCH * M2 * 2;   //     4,096
    float* h2     = ws + o;  o += (size_t)BATCH * M2 * 128; //   262,144
    unsigned* gf_enc = (unsigned*)(ws + o);                 //     4,096

    // 1) hidden = tanh(x @ W1 + b1)
    hidden_kernel<<<(M * 64 + 255) / 256, 256, 0, stream>>>(x, lf_W1, lf_b1,
                                                            hidden, M);
    // 2) lf = tanh(hidden @ W2 + b2)   -> d_out (also the A-matrix for t1)
    gemm_wmma_kernel<1, 64, 0, 0, 64><<<dim3(M / 16, 1), 32, 0, stream>>>(
        hidden, nullptr, nullptr, lf_W2, 64, lf_b2, lf_out, nullptr, 0);

    // 3) FPS chains (stream-serialized; fps2 consumes ctr1)
    fps_kernel<<<BATCH, 256, 0, stream>>>(x, NPTS, M1, ctr1);
    fps_kernel<<<BATCH, 256, 0, stream>>>(ctr1, M1, M2, ctr2);

    // 4) t1 = [lf(64) | zone(1) | pos(2)] @ c1_W (K=67 -> padded 68)
    gemm_wmma_kernel<0, 64, 1, 2, 67><<<dim3(M / 16, 1), 32, 0, stream>>>(
        lf_out, zones, x, c1_W, 64, nullptr, t1, nullptr, 0);

    // 5) SA1 radius-max aggregation (r=0.5, F=64)
    agg_kernel<2><<<(BATCH * M1 * 32) / 256, 256, 0, stream>>>(
        t1, x, ctr1, c1_W + 65 * 64, c1_b, h1, NPTS, M1, 0.25f);

    // 6) t2 = [h1(64) | ctr1(2)] @ c2_W (K=66 -> padded 68)
    gemm_wmma_kernel<0, 64, 2, 0, 66>
        <<<dim3(BATCH * M1 / 16, 2), 32, 0, stream>>>(
            h1, ctr1, nullptr, c2_W, 128, nullptr, t2, nullptr, 0);

    // 7) SA2 radius-max aggregation (r=1.0, F=128)
    agg_kernel<4><<<(BATCH * M2 * 32) / 256, 256, 0, stream>>>(
        t2, ctr1, ctr2, c2_W + 64 * 128, c2_b, h2, M1, M2, 1.0f);

    // 8) gf = rowmax_per_batch([h2(128)|ctr2(2)] @ c3_W + c3_b)
    init_gf_kernel<<<(BATCH * GF + 255) / 256, 256, 0, stream>>>(gf_enc,
                                                                 BATCH * GF);
    gemm_wmma_kernel<2, 128, 2, 0, 130>
        <<<dim3(BATCH * M2 / 16, GF / 64), 32, 0, stream>>>(
            h2, ctr2, nullptr, c3_W, GF, c3_b, nullptr, gf_enc, M2);
    decode_gf_kernel<<<(BATCH * GF + 255) / 256, 256, 0, stream>>>(gf_enc,
                                                                   gf_out,
                                                                   BATCH * GF);
}